// EnergyRefinement_58196806861418
// MI455X (gfx1250) — compile-verified
//
#include <hip/hip_runtime.h>

#define LN 4096
#define IDEAL_C1 6.0f
#define MIN_D 3.4f
#define TARGET_D 9.0f   // 1.5 * IDEAL_C1

typedef __attribute__((ext_vector_type(2))) float v2f;
typedef __attribute__((ext_vector_type(8))) float v8f;

// ---------------------------------------------------------------------------
// Kernel 1: bond energy partial sum (single block, deterministic tree).
// ---------------------------------------------------------------------------
__global__ __launch_bounds__(256)
void er_bond_kernel(const float* __restrict__ coords,
                    float* __restrict__ bond_sum) {
    __shared__ float s[256];
    const int tid = threadIdx.x;
    float acc = 0.0f;
    for (int k = tid; k < LN - 1; k += 256) {
        const float dx = coords[(k + 1) * 3 + 0] - coords[k * 3 + 0];
        const float dy = coords[(k + 1) * 3 + 1] - coords[k * 3 + 1];
        const float dz = coords[(k + 1) * 3 + 2] - coords[k * 3 + 2];
        const float bd = __builtin_amdgcn_sqrtf(fmaf(dx, dx, fmaf(dy, dy, dz * dz)));
        const float t = bd - IDEAL_C1;
        acc = fmaf(t, t, acc);
    }
    s[tid] = acc;
    __syncthreads();
    for (int w = 128; w > 0; w >>= 1) {
        if (tid < w) s[tid] += s[tid + w];
        __syncthreads();
    }
    if (tid == 0) bond_sum[0] = s[0];
}

// ---------------------------------------------------------------------------
// Kernel 2: all-pairs energies. One v_wmma_f32_16x16x4_f32 per 16x16 tile
// computes the squared-distance tile DIRECTLY:
//     D = (-2*coords_rows) x (coords_cols) + (r2_i + r2_j)  =  d^2
// Epilogue is fully branch-free (cndmask only), sqrt is native v_sqrt_f32,
// and the 64MB contact-map stream uses non-temporal loads.
// grid.x = 256 tile-rows; 256 threads = 8 waves; wave w owns 32 col-tiles.
// ---------------------------------------------------------------------------
__global__ __launch_bounds__(256)
void er_pair_kernel(const float* __restrict__ coords,
                    const float* __restrict__ cmap,
                    float* __restrict__ part_clash,
                    float* __restrict__ part_pair,
                    float* __restrict__ part_np) {
    __shared__ v2f   s_ab[2 * LN];      // 64 KB: [2n]={x,y}, [2n+1]={z,0}
    __shared__ float s_r2[LN];          // 16 KB
    __shared__ float s_red[3 * 256];    //  3 KB

    const int tid = threadIdx.x;
    // Stage WMMA operands + squared norms (coords are L2-resident, 48 KB).
    for (int k = tid; k < LN; k += 256) {
        const float x = coords[k * 3 + 0];
        const float y = coords[k * 3 + 1];
        const float z = coords[k * 3 + 2];
        v2f lo; lo.x = x;    lo.y = y;
        v2f hi; hi.x = z;    hi.y = 0.0f;
        s_ab[2 * k + 0] = lo;
        s_ab[2 * k + 1] = hi;
        s_r2[k] = fmaf(x, x, fmaf(y, y, z * z));
    }
    __syncthreads();

    const int ti   = blockIdx.x;     // rows m in [ti*16, ti*16+16)
    const int wave = tid >> 5;
    const int lane = tid & 31;
    const int hv   = lane >> 4;      // half-wave selector (K-split)
    const int r    = lane & 15;

    // A operand (16x4 f32), pre-scaled by -2 so the WMMA emits d^2 directly.
    const int i_row = ti * 16 + r;
    v2f a = s_ab[2 * i_row + hv];
    a.x *= -2.0f;
    a.y *= -2.0f;

    // Per-lane output rows m (fixed across the tj loop) and their r2.
    int   mrow[8];
    float r2i[8];
#pragma unroll
    for (int g = 0; g < 8; ++g) {
        mrow[g] = ti * 16 + g + 8 * hv;
        r2i[g]  = s_r2[mrow[g]];
    }

    float acc_clash = 0.0f, acc_pair = 0.0f, acc_np = 0.0f;

    for (int t = 0; t < 32; ++t) {
        const int tj = wave * 32 + t;
        const int n  = tj * 16 + r;              // output column this lane owns

        const v2f   b   = s_ab[2 * n + hv];      // one ds_load_b64
        const float r2j = s_r2[n];

        // Accumulator carries r2_i + r2_j, so D = r2_i + r2_j - 2*(ci.cj) = d^2
        v8f c;
#pragma unroll
        for (int g = 0; g < 8; ++g) c[g] = r2i[g] + r2j;

        v8f dd = __builtin_amdgcn_wmma_f32_16x16x4_f32(
            /*neg_a=*/false, a, /*neg_b=*/false, b,
            /*c_mod=*/(short)0, c, /*reuse_a=*/false, /*reuse_b=*/false);

        // 64 MB stream: one contact value per (m,n) element; single-use -> NT
        float cmv[8];
#pragma unroll
        for (int g = 0; g < 8; ++g)
            cmv[g] = __builtin_nontemporal_load(&cmap[(size_t)mrow[g] * LN + n]);

#pragma unroll
        for (int g = 0; g < 8; ++g) {
            const int sep = n - mrow[g];
            const float d = __builtin_amdgcn_sqrtf(fmaxf(dd[g], 0.0f)) + 1e-8f;

            // clash: upper triangle j >= i + 3 (branch-free)
            float tc = fmaxf(MIN_D - d, 0.0f);
            tc = (sep >= 3) ? tc : 0.0f;
            acc_clash = fmaf(tc, tc, acc_clash);

            // contacts: count all; energy only when |sep| >= 3 (branch-free)
            const float isc = (cmv[g] > 0.5f) ? 1.0f : 0.0f;
            acc_np += isc;
            const float tp = d - TARGET_D;
            const float w  = ((unsigned)(sep + 2) > 4u) ? isc : 0.0f;
            acc_pair = fmaf(w * tp, tp, acc_pair);
        }
    }

    // deterministic block reduction
    s_red[tid]       = acc_clash;
    s_red[256 + tid] = acc_pair;
    s_red[512 + tid] = acc_np;
    __syncthreads();
    for (int w = 128; w > 0; w >>= 1) {
        if (tid < w) {
            s_red[tid]       += s_red[tid + w];
            s_red[256 + tid] += s_red[256 + tid + w];
            s_red[512 + tid] += s_red[512 + tid + w];
        }
        __syncthreads();
    }
    if (tid == 0) {
        part_clash[blockIdx.x] = s_red[0];
        part_pair[blockIdx.x]  = s_red[256];
        part_np[blockIdx.x]    = s_red[512];
    }
}

// ---------------------------------------------------------------------------
// Kernel 3: combine partials into the scalar output.
// ---------------------------------------------------------------------------
__global__ __launch_bounds__(256)
void er_finalize_kernel(const float* __restrict__ part_clash,
                        const float* __restrict__ part_pair,
                        const float* __restrict__ part_np,
                        const float* __restrict__ bond_sum,
                        float* __restrict__ out) {
    __shared__ float s[3 * 256];
    const int tid = threadIdx.x;
    s[tid]       = part_clash[tid];
    s[256 + tid] = part_pair[tid];
    s[512 + tid] = part_np[tid];
    __syncthreads();
    for (int w = 128; w > 0; w >>= 1) {
        if (tid < w) {
            s[tid]       += s[tid + w];
            s[256 + tid] += s[256 + tid + w];
            s[512 + tid] += s[512 + tid + w];
        }
        __syncthreads();
    }
    if (tid == 0) {
        const float e_bond  = bond_sum[0] / (float)(LN - 1);
        const float e_clash = s[0] / (float)LN;
        const float npair   = fmaxf(s[512], 1.0f);
        const float e_pair  = s[256] / npair;
        out[0] = 1.0f * e_bond + 2.0f * e_clash + 0.5f * e_pair;
    }
}

// ---------------------------------------------------------------------------
extern "C" void kernel_launch(void* const* d_in, const int* in_sizes, int n_in,
                              void* d_out, int out_size, void* d_ws, size_t ws_size,
                              hipStream_t stream) {
    const float* coords = (const float*)d_in[0];   // [4096, 3] f32
    const float* cmap   = (const float*)d_in[1];   // [4096, 4096] f32

    float* ws         = (float*)d_ws;
    float* part_clash = ws;                 // 256
    float* part_pair  = ws + 256;           // 256
    float* part_np    = ws + 512;           // 256
    float* bond_sum   = ws + 768;           // 1

    er_bond_kernel<<<1, 256, 0, stream>>>(coords, bond_sum);
    er_pair_kernel<<<256, 256, 0, stream>>>(coords, cmap,
                                            part_clash, part_pair, part_np);
    er_finalize_kernel<<<1, 256, 0, stream>>>(part_clash, part_pair, part_np,
                                              bond_sum, (float*)d_out);
}